// PhenotypeAwareEncoder_45251775431079
// MI455X (gfx1250) — compile-verified
//
#include <hip/hip_runtime.h>
#include <math.h>

#define NG 2000
#define NC 50000
#define NP 64
#define NT 52064
#define NE 1200000
#define D  128

typedef __attribute__((ext_vector_type(2))) float v2f;
typedef __attribute__((ext_vector_type(8))) float v8f;

__device__ __forceinline__ float celuf(float x) { return x > 0.f ? x : expm1f(x); }

__device__ __forceinline__ void atomic_add_f32(float* p, float v) {
  __hip_atomic_fetch_add(p, v, __ATOMIC_RELAXED, __HIP_MEMORY_SCOPE_AGENT);
}

// Monotonic u32 key so float segment-max can use integer atomicMax exactly.
__device__ __forceinline__ unsigned fkey(float f) {
  unsigned u = __float_as_uint(f);
  return (u & 0x80000000u) ? ~u : (u | 0x80000000u);
}
__device__ __forceinline__ float fdec(unsigned k) {
  unsigned u = (k & 0x80000000u) ? (k & 0x7fffffffu) : ~k;
  return __uint_as_float(u);
}

__global__ void fill_f32(float* p, float v, int n) {
  int i = blockIdx.x * blockDim.x + threadIdx.x;
  if (i < n) p[i] = v;
}

__global__ void copy_f32(const float* __restrict__ s, float* __restrict__ d, int n) {
  int i = blockIdx.x * blockDim.x + threadIdx.x;
  if (i < n) d[i] = s[i];
}

// ---------------- WMMA fp32 GEMM: C[M x N] = act(A[M x 128] @ W[128 x N] + bias[N])
// Register-blocked: one wave owns a 64x16 tile = 4 independent v8f accumulators.
// B fragment loaded once per K-step, reused 4x; 4 independent WMMA chains hide the
// WMMA->WMMA RAW hazard. blockDim.x must be 2*N. Tail blocks clamp A-row loads
// (EXEC stays all-ones for WMMA) and predicate stores wave-uniformly.
template <int ACT>
__global__ __launch_bounds__(256) void gemm128(const float* __restrict__ A,
                                               const float* __restrict__ W,
                                               const float* __restrict__ bias,
                                               float* __restrict__ C, int N, int M) {
  const int wave = threadIdx.x >> 5;
  const int lane = threadIdx.x & 31;
  const int col0 = wave << 4;
  const int row0 = blockIdx.x << 6;   // 64 rows per block
  const int half = lane >> 4;         // 0: lanes 0-15, 1: lanes 16-31
  const int m    = lane & 15;
  const int n    = col0 + m;          // this lane's output column (B/C/D stripe rule)

  const float* Ar[4];
#pragma unroll
  for (int t = 0; t < 4; ++t) {
    int r = row0 + 16 * t + m;
    if (r > M - 1) r = M - 1;         // clamp: tail subtiles read duplicate rows
    Ar[t] = A + (size_t)r * D;
  }
  const float* Wp = W + n;

  v8f acc[4] = {{}, {}, {}, {}};
#pragma unroll 2
  for (int k = 0; k < D; k += 4) {
    const int ka = k + 2 * half;      // A 16x4 layout: v0={K0|K2}, v1={K1|K3}
    v2f b;
    b.x = Wp[(size_t)ka * N];         // B 4x16 layout mirrors A across rows of W
    b.y = Wp[(size_t)(ka + 1) * N];
#pragma unroll
    for (int t = 0; t < 4; ++t) {
      v2f a;
      a.x = Ar[t][ka];
      a.y = Ar[t][ka + 1];
      acc[t] = __builtin_amdgcn_wmma_f32_16x16x4_f32(false, a, false, b, (short)0,
                                                     acc[t], false, false);
    }
  }

  const float bv = bias ? bias[n] : 0.f;
#pragma unroll
  for (int t = 0; t < 4; ++t) {
    const int rb = row0 + 16 * t;
    if (rb < M) {                     // wave-uniform store predicate
#pragma unroll
      for (int v = 0; v < 8; ++v) {   // D layout: vgpr v -> row v (+8 for upper half)
        float x = acc[t][v] + bv;
        if (ACT == 1) x = fmaxf(x, 0.f);
        if (ACT == 2) x = celuf(x);
        C[(size_t)(rb + v + 8 * half) * N + n] = x;
      }
    }
  }
}

// ---------------- pheno MLP layer 1: T[64x128] = celu(pheno_raw * W1 + b1)
__global__ void pheno_l1(const float* __restrict__ pr, const float* __restrict__ W1,
                         const float* __restrict__ b1, float* __restrict__ T) {
  int i = blockIdx.x * blockDim.x + threadIdx.x;
  if (i < NP * D) {
    int r = i >> 7, c = i & (D - 1);
    T[i] = celuf(pr[r] * W1[c] + b1[c]);
  }
}

// ---------------- per-node attention coefficients: a_s = h.a_src, a_d = h.a_dst
__global__ void attn_scores(const float* __restrict__ H, const float* __restrict__ av,
                            const float* __restrict__ bv, float* __restrict__ os,
                            float* __restrict__ od, int n) {
  int wid  = (blockIdx.x * blockDim.x + threadIdx.x) >> 5;
  int lane = threadIdx.x & 31;
  if (wid >= n) return;
  float4 h = ((const float4*)(H + (size_t)wid * D))[lane];
  float4 a = ((const float4*)av)[lane];
  float4 b = ((const float4*)bv)[lane];
  float ps = h.x * a.x + h.y * a.y + h.z * a.z + h.w * a.w;
  float pd = h.x * b.x + h.y * b.y + h.z * b.z + h.w * b.w;
  for (int o = 16; o; o >>= 1) {
    ps += __shfl_xor(ps, o, 32);
    pd += __shfl_xor(pd, o, 32);
  }
  if (lane == 0) { os[wid] = ps; od[wid] = pd; }
}

// ---------------- edge pass 1: logits + segment-max (u32-keyed)
__global__ void edge_logits(const int* __restrict__ src, const int* __restrict__ dst,
                            const float* __restrict__ as_, const float* __restrict__ ad_,
                            float* __restrict__ ebuf, unsigned* __restrict__ emax) {
  int e = blockIdx.x * blockDim.x + threadIdx.x;
  if (e >= NE) return;
  float v = as_[src[e]] + ad_[dst[e]];
  v = v > 0.f ? v : 0.2f * v;           // leaky_relu, slope 0.2
  ebuf[e] = v;
  atomicMax(&emax[dst[e]], fkey(v));
}

// decode keys in-place; non-finite (empty segment) -> 0 (matches reference)
__global__ void emax_fix(float* emax, int n) {
  int i = blockIdx.x * blockDim.x + threadIdx.x;
  if (i >= n) return;
  float m = fdec(((unsigned*)emax)[i]);
  if (!(fabsf(m) < INFINITY)) m = 0.f;
  emax[i] = m;
}

// ---------------- edge pass 2: num = exp(e - max[dst]); den[dst] += num
__global__ void edge_softmax_num(const int* __restrict__ dst, const float* __restrict__ emax,
                                 float* __restrict__ ebuf, float* __restrict__ den) {
  int e = blockIdx.x * blockDim.x + threadIdx.x;
  if (e >= NE) return;
  int d = dst[e];
  float nu = __expf(ebuf[e] - emax[d]);
  ebuf[e] = nu;
  atomic_add_f32(&den[d], nu);
}

// ---------------- edge pass 3: out[dst] += alpha * h[src]  (wave per edge, float4/lane)
__global__ void edge_aggregate(const int* __restrict__ src, const int* __restrict__ dst,
                               const float* __restrict__ num, const float* __restrict__ den,
                               const float* __restrict__ H, float* __restrict__ out) {
  int wid  = (blockIdx.x * blockDim.x + threadIdx.x) >> 5;
  int lane = threadIdx.x & 31;
  if (wid >= NE) return;
  int s = src[wid], d = dst[wid];
  float alpha = num[wid] / fmaxf(den[d], 1e-16f);
  float4 h = ((const float4*)(H + (size_t)s * D))[lane];   // L2-resident gather
  float* o = out + (size_t)d * D + lane * 4;
  atomic_add_f32(o + 0, alpha * h.x);
  atomic_add_f32(o + 1, alpha * h.y);
  atomic_add_f32(o + 2, alpha * h.z);
  atomic_add_f32(o + 3, alpha * h.w);
}

// ---------------- conv epilogue: out = celu(acc + b[col])
__global__ void conv_finalize(const float* acc, const float* __restrict__ b, float* out, int n) {
  int i = blockIdx.x * blockDim.x + threadIdx.x;
  if (i < n) out[i] = celuf(acc[i] + b[i & (D - 1)]);
}

// ---------------- degree count per sample
__global__ void deg_count(const int* __restrict__ c2s, float* __restrict__ deg) {
  int i = blockIdx.x * blockDim.x + threadIdx.x;
  if (i < NC) atomic_add_f32(&deg[c2s[i]], 1.f);
}

// ---------------- per-sample precompute: zWP[s] = (z_ph[s]@WP)/dnorm, q[s] = (z_ph[s].wg_hi)/dnorm
__global__ void sample_prep(const float* __restrict__ zp, const float* __restrict__ WP,
                            const float* __restrict__ wg, const float* __restrict__ deg,
                            float* __restrict__ zWP, float* __restrict__ q) {
  int s = blockIdx.x, j = threadIdx.x;
  __shared__ float red[D];
  float dn = sqrtf(fmaxf(deg[s], 1.f));
  const float* z = zp + (size_t)s * D;
  float acc = 0.f;
  for (int k = 0; k < D; ++k) acc += z[k] * WP[(size_t)k * D + j];
  zWP[(size_t)s * D + j] = acc / dn;
  red[j] = z[j] * wg[D + j];
  __syncthreads();
  for (int st = 64; st; st >>= 1) {
    if (j < st) red[j] += red[j + st];
    __syncthreads();
  }
  if (j == 0) q[s] = red[0] / dn;
}

// ---------------- per-cell gate + injection (wave per cell); writes h_inj into x_out slot
__global__ void cell_gate(const float* __restrict__ xloc, const float* __restrict__ zWP,
                          const float* __restrict__ q, const int* __restrict__ c2s,
                          const float* __restrict__ wg, const float* __restrict__ bg,
                          float* __restrict__ inj, float* __restrict__ gout) {
  int wid  = (blockIdx.x * blockDim.x + threadIdx.x) >> 5;
  int lane = threadIdx.x & 31;
  if (wid >= NC) return;
  float4 h = ((const float4*)(xloc + (size_t)(NG + wid) * D))[lane];
  float4 w = ((const float4*)wg)[lane];                    // wg_lo
  float p = h.x * w.x + h.y * w.y + h.z * w.z + h.w * w.w;
  for (int o = 16; o; o >>= 1) p += __shfl_xor(p, o, 32);  // all lanes get full sum
  int s = c2s[wid];
  float g = 1.f / (1.f + __expf(-(p + q[s] + bg[0])));
  float4 zw = ((const float4*)(zWP + (size_t)s * D))[lane];
  float4 o4;
  o4.x = h.x + g * zw.x; o4.y = h.y + g * zw.y;
  o4.z = h.z + g * zw.z; o4.w = h.w + g * zw.w;
  ((float4*)(inj + (size_t)wid * D))[lane] = o4;
  if (lane == 0) gout[wid] = g;
}

extern "C" void kernel_launch(void* const* d_in, const int* in_sizes, int n_in,
                              void* d_out, int out_size, void* d_ws, size_t ws_size,
                              hipStream_t stream) {
  const float* x        = (const float*)d_in[0];
  const int*   ei       = (const int*)d_in[1];
  const int*   e_src    = ei;
  const int*   e_dst    = ei + NE;
  const float* pheno_raw = (const float*)d_in[2];
  const int*   c2s      = (const int*)d_in[3];
  const float* Wg_gene  = (const float*)d_in[4];
  const float* bg_gene  = (const float*)d_in[5];
  const float* Wg_cell  = (const float*)d_in[6];
  const float* bg_cell  = (const float*)d_in[7];
  const float* convW[2]  = {(const float*)d_in[8],  (const float*)d_in[12]};
  const float* convAS[2] = {(const float*)d_in[9],  (const float*)d_in[13]};
  const float* convAD[2] = {(const float*)d_in[10], (const float*)d_in[14]};
  const float* convB[2]  = {(const float*)d_in[11], (const float*)d_in[15]};
  const float* mlp_W1   = (const float*)d_in[16];
  const float* mlp_b1   = (const float*)d_in[17];
  const float* mlp_W2   = (const float*)d_in[18];
  const float* mlp_b2   = (const float*)d_in[19];
  const float* WP       = (const float*)d_in[20];
  const float* Wg_gate  = (const float*)d_in[21];
  const float* bg_gate  = (const float*)d_in[22];
  const float* Wsen     = (const float*)d_in[23];
  const float* bsen     = (const float*)d_in[24];
  const float* Wctx     = (const float*)d_in[25];
  const float* bctx     = (const float*)d_in[26];

  // workspace layout (floats)
  float* ws   = (float*)d_ws;
  float* B0   = ws;                          // x_new -> (reused) conv2 accum -> x_local
  float* B1   = B0 + (size_t)NT * D;         // h_lin per conv (and pheno temp)
  float* B2   = B1 + (size_t)NT * D;         // conv1 accum -> h
  float* AS   = B2 + (size_t)NT * D;
  float* AD   = AS + NT;
  float* EMAX = AD + NT;
  float* DEN  = EMAX + NT;
  float* EBUF = DEN + NT;
  float* DEG  = EBUF + NE;
  float* Q    = DEG + NP;
  float* ZWP  = Q + NP;

  // output layout (floats, tuple order)
  float* XOUT = (float*)d_out;
  float* ZSEN = XOUT + (size_t)NT * D;
  float* ZCTX = ZSEN + (size_t)NC * D;
  float* GOUT = ZCTX + (size_t)NC * 64;
  float* ZPH  = GOUT + NC;

  auto cdiv = [](int a, int b) { return (a + b - 1) / b; };

  // ---- Phase A: node feature embeddings
  gemm128<1><<<cdiv(NG, 64), 256, 0, stream>>>(x, Wg_gene, bg_gene, B0, D, NG);
  gemm128<1><<<cdiv(NC, 64), 256, 0, stream>>>(x + (size_t)NG * D, Wg_cell, bg_cell,
                                               B0 + (size_t)NG * D, D, NC);
  pheno_l1<<<cdiv(NP * D, 256), 256, 0, stream>>>(pheno_raw, mlp_W1, mlp_b1, B1);
  gemm128<2><<<cdiv(NP, 64), 256, 0, stream>>>(B1, mlp_W2, mlp_b2,
                                               B0 + (size_t)(NG + NC) * D, D, NP);
  copy_f32<<<cdiv(NP * D, 256), 256, 0, stream>>>(B0 + (size_t)(NG + NC) * D, ZPH, NP * D);

  // ---- Phase B: two GATConv layers
  float* convIn[2]  = {B0, B2};
  float* convOut[2] = {B2, B0};
  for (int c = 0; c < 2; ++c) {
    gemm128<0><<<cdiv(NT, 64), 256, 0, stream>>>(convIn[c], convW[c], nullptr, B1, D, NT);
    attn_scores<<<cdiv(NT * 32, 256), 256, 0, stream>>>(B1, convAS[c], convAD[c], AS, AD, NT);
    fill_f32<<<cdiv(NT, 256), 256, 0, stream>>>(EMAX, 0.f, NT);         // key 0 == "-inf"
    fill_f32<<<cdiv(NT, 256), 256, 0, stream>>>(DEN, 0.f, NT);
    fill_f32<<<cdiv(NT * D, 256), 256, 0, stream>>>(convOut[c], 0.f, NT * D);
    edge_logits<<<cdiv(NE, 256), 256, 0, stream>>>(e_src, e_dst, AS, AD, EBUF,
                                                   (unsigned*)EMAX);
    emax_fix<<<cdiv(NT, 256), 256, 0, stream>>>(EMAX, NT);
    edge_softmax_num<<<cdiv(NE, 256), 256, 0, stream>>>(e_dst, EMAX, EBUF, DEN);
    edge_aggregate<<<cdiv(NE * 32, 256), 256, 0, stream>>>(e_src, e_dst, EBUF, DEN, B1,
                                                           convOut[c]);
    conv_finalize<<<cdiv(NT * D, 256), 256, 0, stream>>>(convOut[c], convB[c],
                                                         convOut[c], NT * D);
  }
  // B0 now holds x_local

  // ---- Phase C: phenotype injection + heads
  fill_f32<<<1, 64, 0, stream>>>(DEG, 0.f, NP);
  deg_count<<<cdiv(NC, 256), 256, 0, stream>>>(c2s, DEG);
  sample_prep<<<NP, D, 0, stream>>>(ZPH, WP, Wg_gate, DEG, ZWP, Q);
  cell_gate<<<cdiv(NC * 32, 256), 256, 0, stream>>>(B0, ZWP, Q, c2s, Wg_gate, bg_gate,
                                                    XOUT + (size_t)NG * D, GOUT);
  gemm128<2><<<cdiv(NC, 64), 256, 0, stream>>>(XOUT + (size_t)NG * D, Wsen, bsen, ZSEN, D, NC);
  gemm128<2><<<cdiv(NC, 64), 128, 0, stream>>>(XOUT + (size_t)NG * D, Wctx, bctx, ZCTX, 64, NC);

  // remaining x_out rows (genes + pheno) from x_local
  copy_f32<<<cdiv(NG * D, 256), 256, 0, stream>>>(B0, XOUT, NG * D);
  copy_f32<<<cdiv(NP * D, 256), 256, 0, stream>>>(B0 + (size_t)(NG + NC) * D,
                                                  XOUT + (size_t)(NG + NC) * D, NP * D);
}